// LSR_75067438400064
// MI455X (gfx1250) — compile-verified
//
#include <hip/hip_runtime.h>

typedef __attribute__((ext_vector_type(4))) float v4f;

#define NCLASS 80
#define BLOCK 256
#define NBLOCKS 2560
#define PRIOR_F4 ((NCLASS * NCLASS) / 4)   // 6400 floats / 4 = 1600 b128 copies

// ---------------------------------------------------------------------------
// Kernel A: stream input, gather prior rows from LDS, per-block partial sums.
// ---------------------------------------------------------------------------
__global__ __launch_bounds__(BLOCK) void lsr_partial_kernel(
    const float* __restrict__ inp,      // [B, 80] probabilities
    const int* __restrict__ target,     // [B] class ids
    const float* __restrict__ prior,    // [80, 80]
    float* __restrict__ partial,        // [gridDim.x] partial sums of log2(x)*w
    int n4,                             // B*80/4 float4 elements
    int nblocks)
{
    // 16B alignment: required for async b128 LDS writes and ds_load_b128 reads
    __shared__ __align__(16) float sPrior[NCLASS * NCLASS];   // 25.6 KB
    __shared__ float sRed[BLOCK];

    // ---- CDNA5 async copy: priorDis -> LDS (1600 x global_load_async_to_lds_b128)
    {
        const int tid = threadIdx.x;
        for (int k = tid; k < PRIOR_F4; k += BLOCK) {
            // low 32 bits of a flat shared-pointer are the LDS byte offset
            unsigned lds_off = (unsigned)(unsigned long long)(&sPrior[k * 4]);
            const float* gsrc = prior + k * 4;
            asm volatile("global_load_async_to_lds_b128 %0, %1, off"
                         :
                         : "v"(lds_off), "v"(gsrc)
                         : "memory");
        }
        asm volatile("s_wait_asynccnt 0" ::: "memory");
    }
    __syncthreads();

    // ---- streaming main loop: grid-stride over float4s ----
    const v4f* __restrict__ inp4 = (const v4f*)inp;
    const int stride = nblocks * BLOCK;
    float acc = 0.0f;

    for (int f4 = blockIdx.x * BLOCK + threadIdx.x; f4 < n4; f4 += stride) {
        // prefetch next stride's cacheline (speculative)
        __builtin_prefetch(&inp4[f4 + stride], 0, 0);

        // touch-once data: non-temporal b128 load
        v4f x = __builtin_nontemporal_load(&inp4[f4]);

        const int row = f4 / (NCLASS / 4);          // f4 / 20
        const int c4  = f4 - row * (NCLASS / 4);    // column group 0..19
        const int t   = target[row];                // hot in cache (20x reuse)

        // 16B-aligned gather from the LDS prior tile (ds_load_b128)
        const v4f w = *(const v4f*)&sPrior[t * NCLASS + c4 * 4];

        // accumulate log2(x)*w; ln2 folded into final scale
        acc += __log2f(x.x) * w.x;
        acc += __log2f(x.y) * w.y;
        acc += __log2f(x.z) * w.z;
        acc += __log2f(x.w) * w.w;
    }

    // ---- deterministic in-block tree reduction ----
    sRed[threadIdx.x] = acc;
    __syncthreads();
#pragma unroll
    for (int s = BLOCK / 2; s > 0; s >>= 1) {
        if (threadIdx.x < s) sRed[threadIdx.x] += sRed[threadIdx.x + s];
        __syncthreads();
    }
    if (threadIdx.x == 0) partial[blockIdx.x] = sRed[0];
}

// ---------------------------------------------------------------------------
// Kernel B: reduce partials in f64, apply -ln2/B, write scalar result.
// ---------------------------------------------------------------------------
__global__ __launch_bounds__(BLOCK) void lsr_final_kernel(
    const float* __restrict__ partial, int n, float* __restrict__ out,
    double neg_ln2_over_B)
{
    __shared__ double sRed[BLOCK];
    double a = 0.0;
    for (int i = threadIdx.x; i < n; i += BLOCK) a += (double)partial[i];
    sRed[threadIdx.x] = a;
    __syncthreads();
#pragma unroll
    for (int s = BLOCK / 2; s > 0; s >>= 1) {
        if (threadIdx.x < s) sRed[threadIdx.x] += sRed[threadIdx.x + s];
        __syncthreads();
    }
    if (threadIdx.x == 0) out[0] = (float)(sRed[0] * neg_ln2_over_B);
}

// ---------------------------------------------------------------------------
extern "C" void kernel_launch(void* const* d_in, const int* in_sizes, int n_in,
                              void* d_out, int out_size, void* d_ws, size_t ws_size,
                              hipStream_t stream) {
    const float* inp    = (const float*)d_in[0];   // [B,80] fp32
    const int*   target = (const int*)d_in[1];     // [B] int
    const float* prior  = (const float*)d_in[2];   // [80,80] fp32
    float*       out    = (float*)d_out;           // scalar fp32

    const int B  = in_sizes[1];
    const int n4 = in_sizes[0] / 4;                // B*80/4

    int nblocks = NBLOCKS;
    if ((size_t)nblocks * sizeof(float) > ws_size)
        nblocks = (int)(ws_size / sizeof(float));
    if (nblocks < 1) nblocks = 1;

    float* partial = (float*)d_ws;

    lsr_partial_kernel<<<nblocks, BLOCK, 0, stream>>>(
        inp, target, prior, partial, n4, nblocks);

    const double neg_ln2_over_B = -0.6931471805599453 / (double)B;
    lsr_final_kernel<<<1, BLOCK, 0, stream>>>(partial, nblocks, out,
                                              neg_ln2_over_B);
}